// Attention_90598040141856
// MI455X (gfx1250) — compile-verified
//
#include <hip/hip_runtime.h>

// ---------------------------------------------------------------------------
// MI455X (gfx1250, wave32) implementation of the 2-branch SR attention.
// All matrix contractions go through one fp32 WMMA NT GEMM
// (v_wmma_f32_16x16x4_f32).  Global->LDS staging uses the CDNA5 async
// copy path (GLOBAL_LOAD_ASYNC_TO_LDS_B128 / ASYNCcnt) with double-buffered
// LDS so HBM traffic overlaps the WMMA work.
// ---------------------------------------------------------------------------

typedef float v2f __attribute__((ext_vector_type(2)));
typedef float v8f __attribute__((ext_vector_type(8)));
typedef int   v4i __attribute__((ext_vector_type(4)));

#define B_   16
#define C_   512
#define C2_  256
#define QN   1024
#define QB   16384      // QN * B_

// ---- CDNA5 async global->LDS copy (ASYNCcnt path) -------------------------
// clang-22 prototype: void(v4i AS1*, v4i AS3*, imm int offset, imm int cpol)
__device__ __forceinline__ void async_copy_b128(void* lds, const void* g)
{
#if __has_builtin(__builtin_amdgcn_global_load_async_to_lds_b128)
    __builtin_amdgcn_global_load_async_to_lds_b128(
        (__attribute__((address_space(1))) v4i*)g,
        (__attribute__((address_space(3))) v4i*)lds,
        0, 0);
#else
    unsigned l = (unsigned)(unsigned long long)
        (__attribute__((address_space(3))) void*)lds;
    asm volatile("global_load_async_to_lds_b128 %0, %1, off"
                 :: "v"(l), "v"(g) : "memory");
#endif
}

__device__ __forceinline__ void wait_async0()
{
#if __has_builtin(__builtin_amdgcn_s_wait_asynccnt)
    __builtin_amdgcn_s_wait_asynccnt(0);
#else
    asm volatile("s_wait_asynccnt 0x0" ::: "memory");
#endif
}

// ======================= WMMA fp32 NT GEMM =================================
// D[m,n] = sum_k A[m,k]*B[n,k] (+bias[n]).  Batched via blockIdx.z with
// (outer,inner) stride decomposition.  Block: 256 threads = 8 wave32,
// wave grid 2(M)x4(N) -> 32x64 block tile, K chunked by 32 through
// double-buffered LDS filled by async copies.
#define KC 32
__global__ __launch_bounds__(256) void gemm_nt(
    const float* __restrict__ A, const float* __restrict__ Bm,
    const float* __restrict__ bias, float* __restrict__ D,
    int K, int lda, int ldb, int ldc, int inner,
    long long sAo, long long sAi, long long sBo, long long sBi,
    long long sCo, long long sCi)
{
    __shared__ float As[2][32][KC + 4];
    __shared__ float Bs[2][64][KC + 4];

    const int z  = blockIdx.z;
    const int zo = z / inner, zi = z % inner;
    A  += zo * sAo + zi * sAi;
    Bm += zo * sBo + zi * sBi;
    D  += zo * sCo + zi * sCi;

    const int t    = threadIdx.x;
    const int lane = t & 31;
    const int wid  = t >> 5;
    const int wm   = wid & 1;            // 0..1 : 16-row tile
    const int wn   = wid >> 1;           // 0..3 : 16-col tile
    const long long rowBase = (long long)blockIdx.y * 32;
    const long long colBase = (long long)blockIdx.x * 64;

    const int la  = t >> 3;              // 0..31 staging row
    const int lc4 = (t & 7) << 2;        // 0..28 staging col (float4)

    // per-thread global source pointers for the three async b128 transfers
    const float* gA  = A  + (rowBase + la) * (long long)lda + lc4;
    const float* gB0 = Bm + (colBase + la) * (long long)ldb + lc4;
    const float* gB1 = Bm + (colBase + la + 32) * (long long)ldb + lc4;

    // ISA fragment layout: lanes 0-15 -> K {0,1}, lanes 16-31 -> K {2,3}
    const int mrow  = wm * 16 + (lane & 15);
    const int nrow  = wn * 16 + (lane & 15);
    const int khalf = (lane >> 4) << 1;  // 0 or 2

    v8f acc = {0.f, 0.f, 0.f, 0.f, 0.f, 0.f, 0.f, 0.f};

    // prologue: stage chunk 0 into buffer 0
    async_copy_b128(&As[0][la][lc4],      gA);
    async_copy_b128(&Bs[0][la][lc4],      gB0);
    async_copy_b128(&Bs[0][la + 32][lc4], gB1);
    wait_async0();
    __syncthreads();

    int buf = 0;
    for (int kk = 0; kk < K; kk += KC) {
        const int nb = buf ^ 1;
        if (kk + KC < K) {   // prefetch next chunk into the other buffer
            async_copy_b128(&As[nb][la][lc4],      gA  + kk + KC);
            async_copy_b128(&Bs[nb][la][lc4],      gB0 + kk + KC);
            async_copy_b128(&Bs[nb][la + 32][lc4], gB1 + kk + KC);
        }
#pragma unroll
        for (int k4 = 0; k4 < KC / 4; ++k4) {
            v2f a = *(const v2f*)&As[buf][mrow][(k4 << 2) + khalf];
            v2f b = *(const v2f*)&Bs[buf][nrow][(k4 << 2) + khalf];
            acc = __builtin_amdgcn_wmma_f32_16x16x4_f32(
                false, a, false, b, (short)0, acc, false, false);
        }
        wait_async0();       // my async writes to buffer nb have landed
        __syncthreads();     // all waves: reads of buf done, writes of nb done
        buf = nb;
    }

    // C/D layout: VGPR r -> row r (lanes 0-15) / row r+8 (lanes 16-31)
    const long long gn = colBase + wn * 16 + (lane & 15);
    const float bv = bias ? bias[gn] : 0.f;
    const long long rb = rowBase + wm * 16 + ((lane >> 4) << 3);
#pragma unroll
    for (int r = 0; r < 8; ++r)
        D[(rb + r) * (long long)ldc + gn] = acc[r] + bv;
}

// ======================= im2col for the SR patch convs =====================
// stride==kernel => zero-redundancy permute.  Also folds in the faithful raw
// reshape: key.transpose(0,2,1).reshape(B,C,H,W) -- gather straight from the
// original (N,B,C) tensor.
__global__ __launch_bounds__(256) void im2col_sr(
    const float* __restrict__ src, float* __restrict__ dst,
    int P, int S, unsigned total)
{
    unsigned idx = blockIdx.x * 256u + threadIdx.x;
    if (idx >= total) return;
    const unsigned Kc = (unsigned)C_ * S * S;
    unsigned r  = idx / Kc, k = idx - r * Kc;
    unsigned b  = r / (unsigned)(P * P), p = r - b * (unsigned)(P * P);
    unsigned ph = p / P, pw = p - ph * P;
    unsigned ss = (unsigned)(S * S);
    unsigned ci = k / ss, kr = k - ci * ss;
    unsigned kh = kr / S, kw = kr - kh * S;
    unsigned y = ph * S + kh, x = pw * S + kw;
    // flat index into the reinterpreted (B,C,H,W) image
    unsigned long long j = (unsigned long long)b * 2097152ull +
                           (unsigned long long)ci * 4096ull + y * 64u + x;
    unsigned n  = (unsigned)(j >> 13);          // / (C_*B_)
    unsigned c0 = (unsigned)((j >> 4) & 511);   // channel in source
    unsigned b0 = (unsigned)(j & 15);           // batch in source
    dst[idx] = src[(unsigned long long)n * 8192ull + b0 * 512u + c0];
}

// ======================= LayerNorm(C=512) + exact GELU, in place ===========
__global__ __launch_bounds__(256) void ln_gelu(
    float* __restrict__ y, const float* __restrict__ w,
    const float* __restrict__ b)
{
    __shared__ float red[256];
    float* p = y + (long long)blockIdx.x * C_;
    const int t = threadIdx.x;
    float x0 = p[t], x1 = p[t + 256];

    red[t] = x0 + x1;
    __syncthreads();
    for (int off = 128; off; off >>= 1) {
        if (t < off) red[t] += red[t + off];
        __syncthreads();
    }
    const float mean = red[0] * (1.f / C_);
    __syncthreads();

    float d0 = x0 - mean, d1 = x1 - mean;
    red[t] = d0 * d0 + d1 * d1;
    __syncthreads();
    for (int off = 128; off; off >>= 1) {
        if (t < off) red[t] += red[t + off];
        __syncthreads();
    }
    const float rstd = rsqrtf(red[0] * (1.f / C_) + 1e-5f);

    float g0 = d0 * rstd * w[t] + b[t];
    float g1 = d1 * rstd * w[t + 256] + b[t + 256];
    p[t]       = 0.5f * g0 * (1.f + erff(g0 * 0.70710678118654752f));
    p[t + 256] = 0.5f * g1 * (1.f + erff(g1 * 0.70710678118654752f));
}

// ======================= depthwise 3x3 local residual ======================
// v: (B, N, 256) flat (head-proj output).  Writes v + dwconv into the
// TRANSPOSED per-(b,h) layout vT[(b*4+h)*64+d][n] for the attn*V NT GEMM.
__global__ __launch_bounds__(256) void local_res(
    const float* __restrict__ v, const float* __restrict__ lw,
    const float* __restrict__ lb, float* __restrict__ vT,
    int hh, int ww)
{
    const int N = hh * ww;
    const unsigned idx = blockIdx.x * 256u + threadIdx.x;
    const unsigned total = (unsigned)(B_ * C2_) * (unsigned)N;
    if (idx >= total) return;
    const int n  = idx % N;
    const int bc = idx / N;
    const int c  = bc % C2_;
    const int b  = bc / C2_;
    const int y0 = n / ww, x0 = n % ww;
    const float* vb = v + (long long)b * N * C2_ + c;

    float acc = lb[c];
#pragma unroll
    for (int dy = -1; dy <= 1; ++dy) {
        const int y = y0 + dy;
        if ((unsigned)y >= (unsigned)hh) continue;
#pragma unroll
        for (int dx = -1; dx <= 1; ++dx) {
            const int x = x0 + dx;
            if ((unsigned)x >= (unsigned)ww) continue;
            acc += lw[c * 9 + (dy + 1) * 3 + (dx + 1)] *
                   vb[(long long)(y * ww + x) * C2_];
        }
    }
    const float out = vb[(long long)n * C2_] + acc;
    const int h = c >> 6, d = c & 63;
    vT[(((long long)b * 4 + h) * 64 + d) * N + n] = out;
}

// ======================= scaled row softmax, in place ======================
__global__ __launch_bounds__(256) void softmax_rows(
    float* __restrict__ att, int cols, float scale)
{
    __shared__ float red[256];
    float* p = att + (long long)blockIdx.x * cols;
    const int t = threadIdx.x;
    const int per = cols >> 8;                  // 1 (N=256) or 4 (N=1024)

    float vals[4];
    float vmax = -3.4e38f;
    for (int i = 0; i < per; ++i) {
        vals[i] = p[t + (i << 8)] * scale;
        vmax = fmaxf(vmax, vals[i]);
    }
    red[t] = vmax;
    __syncthreads();
    for (int off = 128; off; off >>= 1) {
        if (t < off) red[t] = fmaxf(red[t], red[t + off]);
        __syncthreads();
    }
    vmax = red[0];
    __syncthreads();

    float sum = 0.f;
    for (int i = 0; i < per; ++i) {
        vals[i] = __expf(vals[i] - vmax);
        sum += vals[i];
    }
    red[t] = sum;
    __syncthreads();
    for (int off = 128; off; off >>= 1) {
        if (t < off) red[t] += red[t + off];
        __syncthreads();
    }
    const float inv = 1.f / red[0];
    for (int i = 0; i < per; ++i) p[t + (i << 8)] = vals[i] * inv;
}

// ===========================================================================
extern "C" void kernel_launch(void* const* d_in, const int* in_sizes, int n_in,
                              void* d_out, int out_size, void* d_ws,
                              size_t ws_size, hipStream_t stream)
{
    (void)in_sizes; (void)n_in; (void)out_size;

    const float* query = (const float*)d_in[0];
    const float* key   = (const float*)d_in[1];
    const float* value = (const float*)d_in[2];
    const float* q_w   = (const float*)d_in[5];
    const float* sr1_w = (const float*)d_in[6];
    const float* sr1_b = (const float*)d_in[7];
    const float* n1_w  = (const float*)d_in[8];
    const float* n1_b  = (const float*)d_in[9];
    const float* sr2_w = (const float*)d_in[10];
    const float* sr2_b = (const float*)d_in[11];
    const float* n2_w  = (const float*)d_in[12];
    const float* n2_b  = (const float*)d_in[13];
    const float* k1_w  = (const float*)d_in[14];
    // d_in[15] (v1_w) intentionally unused: reference uses k1_w for v1.
    const float* k2_w  = (const float*)d_in[16];
    const float* v2_w  = (const float*)d_in[17];
    const float* lc1_w = (const float*)d_in[18];
    const float* lc1_b = (const float*)d_in[19];
    const float* lc2_w = (const float*)d_in[20];
    const float* lc2_b = (const float*)d_in[21];
    const float* pj_w  = (const float*)d_in[22];
    const float* pj_b  = (const float*)d_in[23];
    float* out = (float*)d_out;
    float* W   = (float*)d_ws;

    // -------- workspace arena (floats) --------
    const size_t oQY  = 0;                           // (16384,512) q proj (flat reinterp)
    const size_t oK1  = oQY  + 16384ull * 512;       // (B*256,256)
    const size_t oK2  = oK1  + 4096ull  * 256;       // (B*1024,256)
    const size_t oV1T = oK2  + 16384ull * 256;       // (B,4,64,256)
    const size_t oV2T = oV1T + 1048576ull;           // (B,4,64,1024)
    const size_t oX   = oV2T + 4194304ull;           // (16384,512) concat x1|x2
    const size_t oC   = oX   + 16384ull * 512;       // big reusable region
    const size_t oIM  = oC;                          // im2col (33.55M)
    const size_t oY1K = oC   + 33554432ull;          // (4096,512)
    const size_t oY1V = oY1K + 2097152ull;
    const size_t oY2K = oY1V + 2097152ull;           // (16384,512)
    const size_t oY2V = oY2K + 8388608ull;
    const size_t oV1R = oY2V + 8388608ull;           // (B*256,256) raw v1
    const size_t oV2R = oV1R + 1048576ull;           // (B*1024,256) raw v2
    const size_t oATT = oC;                          // attn logits (<= 67.11M)
    const size_t needF = oC + 67108864ull;
    if (ws_size < needF * sizeof(float)) return;     // deterministic guard

    auto gemm = [&](const float* A, const float* Bm, const float* bias, float* D,
                    int M, int N, int K, int lda, int ldb, int ldc,
                    int batches, int inner,
                    long long sAo, long long sAi, long long sBo, long long sBi,
                    long long sCo, long long sCi) {
        dim3 g(N / 64, M / 32, batches);
        gemm_nt<<<g, 256, 0, stream>>>(A, Bm, bias, D, K, lda, ldb, ldc, inner,
                                       sAo, sAi, sBo, sBi, sCo, sCi);
    };

    // 1) q = query @ q_w^T  : (16384,512)x(512,512)
    gemm(query, q_w, nullptr, W + oQY, QB, 512, 512, 512, 512, 512,
         1, 1, 0, 0, 0, 0, 0, 0);

    const unsigned imTot = 33554432u;   // both im2col matrices: 33.55M elems
    const int imBlocks = (int)(imTot / 256u);

    // 2) SR1 (4x4 s4) on key and value: im2col -> GEMM -> LN+GELU
    im2col_sr<<<imBlocks, 256, 0, stream>>>(key, W + oIM, 16, 4, imTot);
    gemm(W + oIM, sr1_w, sr1_b, W + oY1K, 4096, 512, 8192, 8192, 8192, 512,
         1, 1, 0, 0, 0, 0, 0, 0);
    ln_gelu<<<4096, 256, 0, stream>>>(W + oY1K, n1_w, n1_b);

    im2col_sr<<<imBlocks, 256, 0, stream>>>(value, W + oIM, 16, 4, imTot);
    gemm(W + oIM, sr1_w, sr1_b, W + oY1V, 4096, 512, 8192, 8192, 8192, 512,
         1, 1, 0, 0, 0, 0, 0, 0);
    ln_gelu<<<4096, 256, 0, stream>>>(W + oY1V, n1_w, n1_b);

    // 3) SR2 (2x2 s2) on key and value
    im2col_sr<<<imBlocks, 256, 0, stream>>>(key, W + oIM, 32, 2, imTot);
    gemm(W + oIM, sr2_w, sr2_b, W + oY2K, 16384, 512, 2048, 2048, 2048, 512,
         1, 1, 0, 0, 0, 0, 0, 0);
    ln_gelu<<<16384, 256, 0, stream>>>(W + oY2K, n2_w, n2_b);

    im2col_sr<<<imBlocks, 256, 0, stream>>>(value, W + oIM, 32, 2, imTot);
    gemm(W + oIM, sr2_w, sr2_b, W + oY2V, 16384, 512, 2048, 2048, 2048, 512,
         1, 1, 0, 0, 0, 0, 0, 0);
    ln_gelu<<<16384, 256, 0, stream>>>(W + oY2V, n2_w, n2_b);

    // 4) head projections (N=256)
    gemm(W + oY1K, k1_w, nullptr, W + oK1,  4096,  256, 512, 512, 512, 256,
         1, 1, 0, 0, 0, 0, 0, 0);
    gemm(W + oY1V, k1_w, nullptr, W + oV1R, 4096,  256, 512, 512, 512, 256,
         1, 1, 0, 0, 0, 0, 0, 0);                       // faithful: k1_w on value_1
    gemm(W + oY2K, k2_w, nullptr, W + oK2,  16384, 256, 512, 512, 512, 256,
         1, 1, 0, 0, 0, 0, 0, 0);
    gemm(W + oY2V, v2_w, nullptr, W + oV2R, 16384, 256, 512, 512, 512, 256,
         1, 1, 0, 0, 0, 0, 0, 0);

    // 5) local residual (depthwise 3x3) + transpose V for the NT GEMM
    local_res<<<4096,  256, 0, stream>>>(W + oV1R, lc1_w, lc1_b, W + oV1T, 16, 16);
    local_res<<<16384, 256, 0, stream>>>(W + oV2R, lc2_w, lc2_b, W + oV2T, 32, 32);

    const float scale = 0.125f;   // hd=64 -> 64^-0.5

    // 6) branch 1: logits (B*4 batched 1024x256x64) -> softmax -> x1
    gemm(W + oQY, W + oK1, nullptr, W + oATT, 1024, 256, 64, 512, 256, 256,
         64, 4,
         /*A*/ 524288ll, 64ll,
         /*B*/ 65536ll,  64ll,
         /*C*/ 1048576ll, 262144ll);
    softmax_rows<<<65536, 256, 0, stream>>>(W + oATT, 256, scale);
    gemm(W + oATT, W + oV1T, nullptr, W + oX, 1024, 64, 256, 256, 256, 512,
         64, 4,
         /*A*/ 1048576ll, 262144ll,
         /*B*/ 65536ll,   16384ll,
         /*C*/ 524288ll,  64ll);

    // 7) branch 2: logits (1024x1024x64) -> softmax -> x2 (concat at col 256)
    gemm(W + oQY + 256, W + oK2, nullptr, W + oATT, 1024, 1024, 64, 512, 256, 1024,
         64, 4,
         /*A*/ 524288ll, 64ll,
         /*B*/ 262144ll, 64ll,
         /*C*/ 4194304ll, 1048576ll);
    softmax_rows<<<65536, 256, 0, stream>>>(W + oATT, 1024, scale);
    gemm(W + oATT, W + oV2T, nullptr, W + oX + 256, 1024, 64, 1024, 1024, 1024, 512,
         64, 4,
         /*A*/ 4194304ll, 1048576ll,
         /*B*/ 262144ll,  65536ll,
         /*C*/ 524288ll,  64ll);

    // 8) output projection with bias
    gemm(W + oX, pj_w, pj_b, out, QB, 512, 512, 512, 512, 512,
         1, 1, 0, 0, 0, 0, 0, 0);
}